// kendallloss_58325655880083
// MI455X (gfx1250) — compile-verified
//
#include <hip/hip_runtime.h>
#include <stdint.h>

// ---------------------------------------------------------------------------
// Kendall loss, N = 16384.
//   full_sum = sum_{i,j} clip(p_i-p_j,-1,1)*clip(t_i-t_j,-1,1)
//   out      = 1 - full_sum / (N*(N-1))
// VALU-bound all-pairs kernel; inputs staged into LDS via the CDNA5 Tensor
// Data Mover (tensor_load_to_lds, TENSORcnt), clamp via v_med3_f32.
// ---------------------------------------------------------------------------

typedef __attribute__((ext_vector_type(4))) unsigned int v4u;
typedef __attribute__((ext_vector_type(8))) int          v8i;
typedef __attribute__((ext_vector_type(4))) int          v4i;

#define JCHUNK 2048          // j-values staged per block (8 KB per array)
#define BLOCK  256           // threads per block (8 wave32)

__device__ __forceinline__ float clamp1(float x) {
    return __builtin_amdgcn_fmed3f(x, -1.0f, 1.0f);   // v_med3_f32
}

__global__ void kendall_init_kernel(double* acc) {
    acc[0] = 0.0;
}

__global__ __launch_bounds__(BLOCK)
void kendall_pair_kernel(const float* __restrict__ predict,
                         const float* __restrict__ target,
                         double* __restrict__ acc) {
    __shared__ float lp[JCHUNK];
    __shared__ float lt[JCHUNK];
    __shared__ float red[BLOCK / 32];

    const int wave = threadIdx.x >> 5;

    // ---- Wave 0: DMA this block's j-chunk of predict/target into LDS ----
    if (wave == 0) {
        const uint64_t jbase = (uint64_t)blockIdx.y * JCHUNK;
        const uint64_t gaP   = (uint64_t)(uintptr_t)(predict + jbase);
        const uint64_t gaT   = (uint64_t)(uintptr_t)(target  + jbase);
        const uint32_t ldsP  = (uint32_t)(uintptr_t)(void*)lp;  // low 32b = LDS offset
        const uint32_t ldsT  = (uint32_t)(uintptr_t)(void*)lt;

        // D# Group 0: [1:0]=count=1, [63:32]=lds_addr, [120:64]=global_addr,
        //             [127:126]=type=2 ("image")
        v4u g0p = { 1u, ldsP, (uint32_t)gaP,
                    ((uint32_t)((gaP >> 32) & 0x01FFFFFFu)) | 0x80000000u };
        v4u g0t = { 1u, ldsT, (uint32_t)gaT,
                    ((uint32_t)((gaT >> 32) & 0x01FFFFFFu)) | 0x80000000u };

        // D# Group 1: data_size=4B (bits 17:16 = 2); tensor_dim0 = JCHUNK,
        // tensor_dim1 = 1, tile_dim0 = JCHUNK, tile_dim1 = 1,
        // tensor_dim0_stride = JCHUNK; everything else 0.
        v8i g1 = { (int)(2u << 16),                 // w0: data_size
                   (int)((uint32_t)JCHUNK << 16),   // w1[31:16] = tensor_dim0[15:0]
                   (int)(1u << 16),                 // w2[31:16] = tensor_dim1[15:0]
                   (int)((uint32_t)JCHUNK << 16),   // w3[31:16] = tile_dim0
                   1,                               // w4[15:0]  = tile_dim1
                   JCHUNK,                          // w5        = tensor_dim0_stride[31:0]
                   0, 0 };
        v4i gz4 = { 0, 0, 0, 0 };                   // groups 2/3 unused (<=2D)
        v8i gz8 = { 0, 0, 0, 0, 0, 0, 0, 0 };       // extra group (6-arg form)

        __builtin_amdgcn_tensor_load_to_lds(g0p, g1, gz4, gz4, gz8, 0);
        __builtin_amdgcn_tensor_load_to_lds(g0t, g1, gz4, gz4, gz8, 0);
        __builtin_amdgcn_s_wait_tensorcnt(0);       // s_wait_tensorcnt 0
    }
    __syncthreads();                                // publish LDS to all waves

    // ---- Each thread: one i, stream 2048 j from LDS (ds_load_b128) ----
    const int i  = blockIdx.x * BLOCK + threadIdx.x;
    const float pi = predict[i];
    const float ti = target[i];

    const float4* lp4 = (const float4*)lp;
    const float4* lt4 = (const float4*)lt;

    float s = 0.0f;
#pragma unroll 4
    for (int j = 0; j < JCHUNK / 4; ++j) {
        const float4 p = lp4[j];
        const float4 t = lt4[j];
        s = fmaf(clamp1(pi - p.x), clamp1(ti - t.x), s);
        s = fmaf(clamp1(pi - p.y), clamp1(ti - t.y), s);
        s = fmaf(clamp1(pi - p.z), clamp1(ti - t.z), s);
        s = fmaf(clamp1(pi - p.w), clamp1(ti - t.w), s);
    }

    // ---- wave32 shuffle reduction, then cross-wave via LDS ----
    for (int off = 16; off > 0; off >>= 1)
        s += __shfl_down(s, off, 32);
    if ((threadIdx.x & 31) == 0) red[wave] = s;
    __syncthreads();

    if (threadIdx.x == 0) {
        float bs = 0.0f;
#pragma unroll
        for (int w = 0; w < BLOCK / 32; ++w) bs += red[w];
        atomicAdd(acc, (double)bs);                 // global_atomic_add_f64
    }
}

__global__ void kendall_finalize_kernel(const double* __restrict__ acc,
                                        float* __restrict__ out, int n) {
    const double denom = (double)n * (double)(n - 1);
    out[0] = (float)(1.0 - acc[0] / denom);
}

extern "C" void kernel_launch(void* const* d_in, const int* in_sizes, int n_in,
                              void* d_out, int out_size, void* d_ws, size_t ws_size,
                              hipStream_t stream) {
    const float* predict = (const float*)d_in[0];
    const float* target  = (const float*)d_in[1];
    float*  out = (float*)d_out;
    double* acc = (double*)d_ws;
    const int n = in_sizes[0];                      // 16384

    kendall_init_kernel<<<1, 1, 0, stream>>>(acc);

    dim3 grid(n / BLOCK, n / JCHUNK);               // 64 x 8 = 512 blocks
    kendall_pair_kernel<<<grid, BLOCK, 0, stream>>>(predict, target, acc);

    kendall_finalize_kernel<<<1, 1, 0, stream>>>(acc, out, n);
}